// DPQNetwork_81423989997931
// MI455X (gfx1250) — compile-verified
//
#include <hip/hip_runtime.h>

// Problem constants (match reference)
#define B_DIM 8192
#define C_DIM 64
#define K_DIM 256
#define D_DIM 64
#define BN_EPS 1e-5f

#define M_PER_WAVE 16
#define WAVES_PER_BLOCK 8
#define M_PER_BLOCK (M_PER_WAVE * WAVES_PER_BLOCK)   // 128 rows per block
#define MBLOCKS (B_DIM / M_PER_BLOCK)                // 64 blocks per channel
#define LDS_STRIDE 68                                // padded row stride (floats), 16B aligned, bank-spread

typedef float v2f __attribute__((ext_vector_type(2)));
typedef float v8f __attribute__((ext_vector_type(8)));

// ---------------------------------------------------------------------------
// Pass 1: fused GEMM (fp32 WMMA) + running max/argmax/min/argmin + Sum/SumSq
// grid = (MBLOCKS, C), block = 256 (8 wave32)
// ---------------------------------------------------------------------------
__global__ __launch_bounds__(256) void dpq_pass1(
    const float* __restrict__ x,          // (B, C, D)
    const float* __restrict__ cent,       // (C, K, D)
    float* __restrict__ partialS,         // (C, MBLOCKS)
    float* __restrict__ partialS2,        // (C, MBLOCKS)
    float* __restrict__ maxv, int* __restrict__ maxi,   // (B*C)
    float* __restrict__ minv, int* __restrict__ mini)   // (B*C)
{
    const int c    = blockIdx.y;
    const int mblk = blockIdx.x;
    const int tid  = threadIdx.x;
    const int lane = tid & 31;
    const int wave = tid >> 5;

    __shared__ float ldsC[K_DIM * LDS_STRIDE];             // ~68 KB, padded
    __shared__ float wS[WAVES_PER_BLOCK], wS2[WAVES_PER_BLOCK];

    // Stage centroids[c] (256x64 fp32) into LDS, float4 at a time (uniform trip count)
    const float4* csrc = (const float4*)(cent + (size_t)c * K_DIM * D_DIM);
    for (int i = tid; i < K_DIM * (D_DIM / 4); i += 256) {
        const int r = i >> 4;            // row, 16 float4 per row
        const int q = i & 15;
        float4 v = csrc[(size_t)r * (D_DIM / 4) + q];
        *(float4*)&ldsC[r * LDS_STRIDE + q * 4] = v;
    }
    __syncthreads();

    const int m0    = mblk * M_PER_BLOCK + wave * M_PER_WAVE;
    const int half  = lane >> 4;          // 0: K pair {0,1}, 1: K pair {2,3}
    const int l16   = lane & 15;
    const int kbase = 2 * half;

    // A fragments: row m0+l16 of X_c, all D=64, as 16 float2 (per 16x16x4 step)
    // Layout: lanes 0-15 -> VGPR0=K0,VGPR1=K1 ; lanes 16-31 -> K2,K3
    v2f afrag[16];
    {
        const float* arow = x + ((size_t)(m0 + l16) * C_DIM + c) * D_DIM;
        #pragma unroll
        for (int t = 0; t < 16; ++t)
            afrag[t] = *(const v2f*)(arow + 4 * t + kbase);
    }

    float rmax[8], rmin[8];
    int   imax[8], imin[8];
    #pragma unroll
    for (int j = 0; j < 8; ++j) {
        rmax[j] = -__builtin_inff(); rmin[j] = __builtin_inff();
        imax[j] = 0; imin[j] = 0;
    }
    float s = 0.f, s2 = 0.f;

    // N-tiles: 16 tiles of 16 columns cover K=256 centroids
    for (int nt = 0; nt < 16; ++nt) {
        v8f acc = {0.f, 0.f, 0.f, 0.f, 0.f, 0.f, 0.f, 0.f};
        // B fragment source: B[k][n] = Cent_c[n][k] -> contiguous float2 per lane
        const float* brow = &ldsC[(nt * 16 + l16) * LDS_STRIDE + kbase];
        #pragma unroll
        for (int t = 0; t < 16; ++t) {
            v2f bfrag = *(const v2f*)(brow + 4 * t);
            acc = __builtin_amdgcn_wmma_f32_16x16x4_f32(
                false, afrag[t], false, bfrag, (short)0, acc, false, false);
        }
        // C/D layout: VGPR j -> row m0 + j + 8*half, col nt*16 + l16.
        // Columns per lane strictly increase with nt, so strict > keeps
        // first-occurrence semantics within a lane.
        const int col = nt * 16 + l16;
        #pragma unroll
        for (int j = 0; j < 8; ++j) {
            const float v = acc[j];
            s += v; s2 += v * v;
            if (v > rmax[j]) { rmax[j] = v; imax[j] = col; }
            if (v < rmin[j]) { rmin[j] = v; imin[j] = col; }
        }
    }

    // Reduce max/min across the 16 lanes sharing each row (xor stays in-half)
    #pragma unroll
    for (int m = 1; m < 16; m <<= 1) {
        #pragma unroll
        for (int j = 0; j < 8; ++j) {
            float ov = __shfl_xor(rmax[j], m, 32);
            int   oi = __shfl_xor(imax[j], m, 32);
            if (ov > rmax[j] || (ov == rmax[j] && oi < imax[j])) { rmax[j] = ov; imax[j] = oi; }
            float uv = __shfl_xor(rmin[j], m, 32);
            int   ui = __shfl_xor(imin[j], m, 32);
            if (uv < rmin[j] || (uv == rmin[j] && ui < imin[j])) { rmin[j] = uv; imin[j] = ui; }
        }
    }
    if (l16 == 0) {
        #pragma unroll
        for (int j = 0; j < 8; ++j) {
            const int row = m0 + j + 8 * half;
            const size_t o = (size_t)row * C_DIM + c;
            maxv[o] = rmax[j]; maxi[o] = imax[j];
            minv[o] = rmin[j]; mini[o] = imin[j];
        }
    }

    // Deterministic Sum / SumSq: wave xor-reduce, then fixed-order over waves
    #pragma unroll
    for (int m = 1; m < 32; m <<= 1) {
        s  += __shfl_xor(s,  m, 32);
        s2 += __shfl_xor(s2, m, 32);
    }
    if (lane == 0) { wS[wave] = s; wS2[wave] = s2; }
    __syncthreads();
    if (tid == 0) {
        float S = 0.f, S2 = 0.f;
        #pragma unroll
        for (int w = 0; w < WAVES_PER_BLOCK; ++w) { S += wS[w]; S2 += wS2[w]; }
        partialS [c * MBLOCKS + mblk] = S;
        partialS2[c * MBLOCKS + mblk] = S2;
    }
}

// ---------------------------------------------------------------------------
// Pass 2: per-channel BN stats -> affine (a, b), deterministic serial sum
// ---------------------------------------------------------------------------
__global__ void dpq_pass2(const float* __restrict__ pS, const float* __restrict__ pS2,
                          const float* __restrict__ gamma, const float* __restrict__ beta,
                          float* __restrict__ chanA, float* __restrict__ chanB)
{
    const int c = threadIdx.x;
    if (c >= C_DIM) return;
    float S = 0.f, S2 = 0.f;
    for (int m = 0; m < MBLOCKS; ++m) { S += pS[c * MBLOCKS + m]; S2 += pS2[c * MBLOCKS + m]; }
    const float invN = 1.0f / ((float)B_DIM * (float)K_DIM);
    const float mean = S * invN;
    float var = S2 * invN - mean * mean;
    var = var > 0.f ? var : 0.f;
    const float a = gamma[c] * rsqrtf(var + BN_EPS);
    chanA[c] = a;
    chanB[c] = beta[c] - a * mean;
}

// ---------------------------------------------------------------------------
// Pass 3: apply affine; pick max branch (a>0), min branch (a<0), or index 0
// ---------------------------------------------------------------------------
__global__ void dpq_pass3(const float* __restrict__ chanA, const float* __restrict__ chanB,
                          const float* __restrict__ maxv, const int* __restrict__ maxi,
                          const float* __restrict__ minv, const int* __restrict__ mini,
                          int* __restrict__ codes, float* __restrict__ mse)
{
    const size_t t = (size_t)blockIdx.x * blockDim.x + threadIdx.x;
    if (t >= (size_t)B_DIM * C_DIM) return;
    const int c = (int)(t % C_DIM);
    const float a = chanA[c], b = chanB[c];
    int code; float val;
    if (a > 0.f)      { code = maxi[t]; val = a * maxv[t] + b; }
    else if (a < 0.f) { code = mini[t]; val = a * minv[t] + b; }
    else              { code = 0;       val = b; }
    codes[t] = code;
    mse[t]   = val;
}

// ---------------------------------------------------------------------------
extern "C" void kernel_launch(void* const* d_in, const int* in_sizes, int n_in,
                              void* d_out, int out_size, void* d_ws, size_t ws_size,
                              hipStream_t stream) {
    const float* x     = (const float*)d_in[0];   // (B, C, D)
    const float* cent  = (const float*)d_in[1];   // (C, K, D)
    const float* gamma = (const float*)d_in[2];   // (C,)
    const float* beta  = (const float*)d_in[3];   // (C,)

    const size_t BC = (size_t)B_DIM * C_DIM;

    // Workspace layout (floats)
    float* ws        = (float*)d_ws;
    float* partialS  = ws;                                   // C*MBLOCKS
    float* partialS2 = partialS  + C_DIM * MBLOCKS;          // C*MBLOCKS
    float* chanA     = partialS2 + C_DIM * MBLOCKS;          // C
    float* chanB     = chanA + C_DIM;                        // C
    float* maxv      = chanB + C_DIM;                        // B*C
    int*   maxi      = (int*)(maxv + BC);                    // B*C
    float* minv      = (float*)(maxi + BC);                  // B*C
    int*   mini      = (int*)(minv + BC);                    // B*C

    // Output layout: codes (int32, B*C) | mse (f32, B*C) | centroids_k (f32, C*K*D)
    int*   codes    = (int*)d_out;
    float* mse      = (float*)d_out + BC;
    float* cent_out = (float*)d_out + 2 * BC;

    dim3 g1(MBLOCKS, C_DIM);
    dpq_pass1<<<g1, 256, 0, stream>>>(x, cent, partialS, partialS2, maxv, maxi, minv, mini);
    dpq_pass2<<<1, 64, 0, stream>>>(partialS, partialS2, gamma, beta, chanA, chanB);
    dpq_pass3<<<(unsigned)((BC + 255) / 256), 256, 0, stream>>>(
        chanA, chanB, maxv, maxi, minv, mini, codes, mse);
    hipMemcpyAsync(cent_out, cent, (size_t)C_DIM * K_DIM * D_DIM * sizeof(float),
                   hipMemcpyDeviceToDevice, stream);
}